// FraudGNN_RL_66486093742317
// MI455X (gfx1250) — compile-verified
//
#include <hip/hip_runtime.h>
#include <math.h>

typedef __attribute__((ext_vector_type(2))) float v2f;
typedef __attribute__((ext_vector_type(8))) float v8f;
typedef int v4i __attribute__((vector_size(16)));

#define HAVE_ASYNC_LDS (__has_builtin(__builtin_amdgcn_global_load_async_to_lds_b128) && \
                        __has_builtin(__builtin_amdgcn_s_wait_asynccnt))

__device__ __forceinline__ void atomAddF(float* p, float v) {
  __hip_atomic_fetch_add(p, v, __ATOMIC_RELAXED, __HIP_MEMORY_SCOPE_AGENT);
}

// ---------------------------------------------------------------------------
// Repack W_p1 (64 x 256) into Wcat (128 x 128):
//   row j<64  : W_p1[j, 0:128]     (source-endpoint half)
//   row j>=64 : W_p1[j-64,128:256] (dest-endpoint half)
// so that P = x @ Wcat^T gives P[:,0:64]=Pa, P[:,64:128]=Pb.
// ---------------------------------------------------------------------------
__global__ void k_repack(const float* __restrict__ Wp1, float* __restrict__ Wcat) {
  int idx = blockIdx.x * blockDim.x + threadIdx.x;
  if (idx >= 128 * 128) return;
  int j = idx >> 7, k = idx & 127;
  Wcat[idx] = (j < 64) ? Wp1[j * 256 + k] : Wp1[(j - 64) * 256 + 128 + k];
}

// ---------------------------------------------------------------------------
// C[N x 128] = A[N x 128] @ W^T  (W: 128x128 row-major [out][in])
// fp32 WMMA 16x16x4. Block = 256 threads = 8 waves, 128 rows per block.
// W is staged into LDS (async-to-LDS when available) and swizzled into
// per-(kstep,jtile) lane fragments for conflict-free ds_load_b64.
// ---------------------------------------------------------------------------
__global__ __launch_bounds__(256) void k_gemm128(const float* __restrict__ A,
                                                 const float* __restrict__ W,
                                                 float* __restrict__ C,
                                                 int nrows) {
  extern __shared__ float smem[];  // [0..16383] swizzled frags, [16384..32767] raw W
  float* sB = smem;
  const int tid = threadIdx.x;

#if HAVE_ASYNC_LDS
  {
    float* sRaw = smem + 16384;
    // Async DMA: 4096 x b128 chunks, 16 per thread, tracked by ASYNCcnt.
    for (int i = tid; i < 4096; i += 256) {
      __builtin_amdgcn_global_load_async_to_lds_b128(
          (__attribute__((address_space(1))) v4i*)(W + i * 4),
          (__attribute__((address_space(3))) v4i*)(sRaw + i * 4),
          0, 0);
    }
    __builtin_amdgcn_s_wait_asynccnt(0);
    __syncthreads();
    // One-shot swizzle raw W -> fragment layout:
    // B[k][n] = W[n][k], n = j*16 + (lane&15), k = ks*4 + (lane>>4)*2 + pair
    for (int idx = tid; idx < 16384; idx += 256) {
      int pair = idx & 1;
      int lane = (idx >> 1) & 31;
      int j    = (idx >> 6) & 7;
      int ks   = idx >> 9;
      int n = j * 16 + (lane & 15);
      int k = ks * 4 + ((lane >> 4) << 1) + pair;
      sB[idx] = sRaw[n * 128 + k];
    }
    __syncthreads();
  }
#else
  // Fallback: cooperative global->register->LDS swizzled fill.
  for (int idx = tid; idx < 16384; idx += 256) {
    int pair = idx & 1;
    int lane = (idx >> 1) & 31;
    int j    = (idx >> 6) & 7;
    int ks   = idx >> 9;
    int n = j * 16 + (lane & 15);
    int k = ks * 4 + ((lane >> 4) << 1) + pair;
    sB[idx] = W[n * 128 + k];
  }
  __syncthreads();
#endif

  const int lane = tid & 31;
  const int wv   = tid >> 5;
  const int m0   = blockIdx.x * 128 + wv * 16;

  // A fragment row for this lane (clamped so tail blocks stay in-bounds;
  // duplicated rows are masked at the store).
  int r  = m0 + (lane & 15);
  int rc = r < nrows ? r : (nrows - 1);
  const float* arow = A + (size_t)rc * 128 + ((lane >> 4) << 1);

  // Prefetch this lane's full A row (global_prefetch_b8; per-lane addresses
  // cover the whole 16x128 stripe of the wave).
  __builtin_prefetch(arow, 0, 2);
  __builtin_prefetch(arow + 32, 0, 2);
  __builtin_prefetch(arow + 64, 0, 2);
  __builtin_prefetch(arow + 96, 0, 2);

  v8f acc[8] = {};
  const v2f* sB2 = (const v2f*)sB;

  // Software pipeline: A fragment loaded one k-step ahead; all 8 B fragments
  // of a k-step batched before the 8 WMMAs.
  v2f a = *(const v2f*)(arow);
#pragma unroll 4
  for (int ks = 0; ks < 32; ++ks) {
    v2f a_next = (ks < 31) ? *(const v2f*)(arow + (ks + 1) * 4) : a;
    v2f bfrag[8];
#pragma unroll
    for (int j = 0; j < 8; ++j) bfrag[j] = sB2[(ks * 8 + j) * 32 + lane];
#pragma unroll
    for (int j = 0; j < 8; ++j) {
      acc[j] = __builtin_amdgcn_wmma_f32_16x16x4_f32(
          /*neg_a=*/false, a, /*neg_b=*/false, bfrag[j],
          /*c_mod=*/(short)0, acc[j], /*reuse_a=*/false, /*reuse_b=*/false);
    }
    a = a_next;
  }

  // D layout: VGPR rr -> row m0 + (lane>>4)*8 + rr, col = jtile*16 + (lane&15)
  const int rowbase = m0 + ((lane >> 4) << 3);
  const int colbase = lane & 15;
#pragma unroll
  for (int j = 0; j < 8; ++j) {
#pragma unroll
    for (int rr = 0; rr < 8; ++rr) {
      int row = rowbase + rr;
      if (row < nrows) C[(size_t)row * 128 + j * 16 + colbase] = acc[j][rr];
    }
  }
}

// ---------------------------------------------------------------------------
// Edge-weight MLP on precomputed projections. One wave per edge.
//   ew[e] = sigmoid( sum_j relu(Pa[s][j]+Pb[d][j]+b_p1[j]) * w_p2[j] + b_p2 )
// ---------------------------------------------------------------------------
__global__ __launch_bounds__(256) void k_edgew(const float* __restrict__ P,
    const long long* __restrict__ src, const long long* __restrict__ dst,
    const float* __restrict__ bp1, const float* __restrict__ wp2,
    const float* __restrict__ bp2, float* __restrict__ ew, int E) {
  int e = blockIdx.x * 8 + (threadIdx.x >> 5);
  if (e >= E) return;
  int lane = threadIdx.x & 31;
  int s = (int)src[e];
  int d = (int)dst[e];
  const float* Ps = P + (size_t)s * 128;
  const float* Pd = P + (size_t)d * 128;
  float h0 = fmaxf(Ps[lane]      + Pd[64 + lane]      + bp1[lane],      0.f) * wp2[lane];
  float h1 = fmaxf(Ps[32 + lane] + Pd[96 + lane]      + bp1[32 + lane], 0.f) * wp2[32 + lane];
  float sum = h0 + h1;
#pragma unroll
  for (int off = 16; off > 0; off >>= 1) sum += __shfl_xor(sum, off, 32);
  if (lane == 0) ew[e] = 1.f / (1.f + __expf(-(sum + bp2[0])));
}

__global__ void k_fill(float* __restrict__ p, float val, int n) {
  int i = blockIdx.x * blockDim.x + threadIdx.x;
  if (i < n) p[i] = val;
}

__global__ void k_degscatter(const long long* __restrict__ dst,
                             const float* __restrict__ ew,
                             float* __restrict__ deg, int E) {
  int e = blockIdx.x * blockDim.x + threadIdx.x;
  if (e < E) atomAddF(deg + (int)dst[e], ew[e]);
}

__global__ void k_dinv(const float* __restrict__ deg, float* __restrict__ dinv, int n) {
  int i = blockIdx.x * blockDim.x + threadIdx.x;
  if (i >= n) return;
  float d = deg[i];
  dinv[i] = d > 0.f ? rsqrtf(d) : 0.f;
}

// agg init with the self-loop term: agg[i][c] = dinv[i]^2 * xl[i][c]
__global__ void k_agg_init(const float* __restrict__ xl, const float* __restrict__ dinv,
                           float* __restrict__ agg, int n /* N*128 */) {
  int i = blockIdx.x * blockDim.x + threadIdx.x;
  if (i >= n) return;
  float di = dinv[i >> 7];
  agg[i] = di * di * xl[i];
}

// One wave per edge: agg[dst] += dinv[s]*ew*dinv[d] * xl[src]  (128 features)
__global__ __launch_bounds__(256) void k_scatter(const float* __restrict__ xl,
    const long long* __restrict__ src, const long long* __restrict__ dst,
    const float* __restrict__ ew, const float* __restrict__ dinv,
    float* __restrict__ agg, int E) {
  int e = blockIdx.x * 8 + (threadIdx.x >> 5);
  if (e >= E) return;
  int lane = threadIdx.x & 31;
  int s = (int)src[e], d = (int)dst[e];
  float coef = dinv[s] * ew[e] * dinv[d];
  const float4 v = *(const float4*)(xl + (size_t)s * 128 + lane * 4);
  float* ap = agg + (size_t)d * 128 + lane * 4;
  atomAddF(ap + 0, coef * v.x);
  atomAddF(ap + 1, coef * v.y);
  atomAddF(ap + 2, coef * v.z);
  atomAddF(ap + 3, coef * v.w);
}

__global__ void k_bias_relu(const float* __restrict__ in, const float* __restrict__ b,
                            float* __restrict__ out, int n) {
  int i = blockIdx.x * blockDim.x + threadIdx.x;
  if (i >= n) return;
  out[i] = fmaxf(in[i] + b[i & 127], 0.f);
}

// Final 128->2 linear + sigmoid. One wave per node.
__global__ __launch_bounds__(256) void k_final(const float* __restrict__ x2,
    const float* __restrict__ Wl, const float* __restrict__ bl,
    float* __restrict__ out, int N) {
  int i = blockIdx.x * 8 + (threadIdx.x >> 5);
  if (i >= N) return;
  int lane = threadIdx.x & 31;
  const float4 xv = *(const float4*)(x2 + (size_t)i * 128 + lane * 4);
  const float4 w0 = *(const float4*)(Wl + lane * 4);
  const float4 w1 = *(const float4*)(Wl + 128 + lane * 4);
  float d0 = xv.x * w0.x + xv.y * w0.y + xv.z * w0.z + xv.w * w0.w;
  float d1 = xv.x * w1.x + xv.y * w1.y + xv.z * w1.z + xv.w * w1.w;
#pragma unroll
  for (int off = 16; off > 0; off >>= 1) {
    d0 += __shfl_xor(d0, off, 32);
    d1 += __shfl_xor(d1, off, 32);
  }
  if (lane == 0) {
    out[(size_t)i * 2 + 0] = 1.f / (1.f + __expf(-(d0 + bl[0])));
    out[(size_t)i * 2 + 1] = 1.f / (1.f + __expf(-(d1 + bl[1])));
  }
}

// ---------------------------------------------------------------------------
extern "C" void kernel_launch(void* const* d_in, const int* in_sizes, int n_in,
                              void* d_out, int out_size, void* d_ws, size_t ws_size,
                              hipStream_t stream) {
  const float*     x   = (const float*)d_in[0];
  const long long* ei  = (const long long*)d_in[1];  // int64, shape (2, E)
  const float*     Wp1 = (const float*)d_in[2];
  const float*     bp1 = (const float*)d_in[3];
  const float*     Wp2 = (const float*)d_in[4];
  const float*     bp2 = (const float*)d_in[5];
  const float*     W1  = (const float*)d_in[6];
  const float*     b1  = (const float*)d_in[7];
  const float*     W2  = (const float*)d_in[8];
  const float*     b2  = (const float*)d_in[9];
  const float*     Wl  = (const float*)d_in[10];
  const float*     bl  = (const float*)d_in[11];

  const int N = in_sizes[0] / 128;
  const int E = in_sizes[1] / 2;
  const long long* src = ei;
  const long long* dst = ei + E;

  // Workspace carve-up (floats)
  float* ws   = (float*)d_ws;
  float* Wcat = ws;                        // 128*128
  float* deg  = Wcat + 128 * 128;          // N
  float* dinv = deg + N;                   // N
  float* ew   = dinv + N;                  // E
  float* B0   = ew + E;                    // N*128 (P / xl scratch)
  float* B1   = B0 + (size_t)N * 128;      // N*128 (agg)
  float* B2   = B1 + (size_t)N * 128;      // N*128 (x1 / x2)

  const int NF = N * 128;
  const int gemmBlocks = (N + 127) / 128;
  const int edgeWaveBlocks = (E + 7) / 8;
  const int nodeWaveBlocks = (N + 7) / 8;
  const size_t gemmLds = 2 * 16384 * sizeof(float);  // 128 KB (raw + swizzled)

  // --- Edge-weight policy MLP (decomposed) ---
  k_repack<<<(128 * 128 + 255) / 256, 256, 0, stream>>>(Wp1, Wcat);
  k_gemm128<<<gemmBlocks, 256, gemmLds, stream>>>(x, Wcat, B0, N);        // P
  k_edgew<<<edgeWaveBlocks, 256, 0, stream>>>(B0, src, dst, bp1, Wp2, bp2, ew, E);

  // --- Degrees / normalization (self-loops contribute weight 1) ---
  k_fill<<<(N + 255) / 256, 256, 0, stream>>>(deg, 1.0f, N);
  k_degscatter<<<(E + 255) / 256, 256, 0, stream>>>(dst, ew, deg, E);
  k_dinv<<<(N + 255) / 256, 256, 0, stream>>>(deg, dinv, N);

  // --- GCN conv 1 ---
  k_gemm128<<<gemmBlocks, 256, gemmLds, stream>>>(x, W1, B0, N);          // xl1
  k_agg_init<<<(NF + 255) / 256, 256, 0, stream>>>(B0, dinv, B1, NF);     // self-loop
  k_scatter<<<edgeWaveBlocks, 256, 0, stream>>>(B0, src, dst, ew, dinv, B1, E);
  k_bias_relu<<<(NF + 255) / 256, 256, 0, stream>>>(B1, b1, B2, NF);      // x1

  // --- GCN conv 2 ---
  k_gemm128<<<gemmBlocks, 256, gemmLds, stream>>>(B2, W2, B0, N);         // xl2
  k_agg_init<<<(NF + 255) / 256, 256, 0, stream>>>(B0, dinv, B1, NF);
  k_scatter<<<edgeWaveBlocks, 256, 0, stream>>>(B0, src, dst, ew, dinv, B1, E);
  k_bias_relu<<<(NF + 255) / 256, 256, 0, stream>>>(B1, b2, B2, NF);      // x2

  // --- Final linear + sigmoid ---
  k_final<<<nodeWaveBlocks, 256, 0, stream>>>(B2, Wl, bl, (float*)d_out, N);
}